// StudentGraphSAGE_38250978738252
// MI455X (gfx1250) — compile-verified
//
#include <hip/hip_runtime.h>

typedef __attribute__((ext_vector_type(2))) float v2f;
typedef __attribute__((ext_vector_type(8))) float v8f;

#define D_FEAT 128

// ---------------------------------------------------------------------------
// 0) zero the accumulator region of the workspace
// ---------------------------------------------------------------------------
__global__ void sage_zero(float* __restrict__ ws, size_t n) {
  size_t i = (size_t)blockIdx.x * blockDim.x + threadIdx.x;
  if (i < n) ws[i] = 0.0f;
}

// ---------------------------------------------------------------------------
// 1) layer-1 scatter: one wave (32 lanes) per edge; each lane handles 4 feats
//    agg1[dst] += x[src];  cnt[dst] += 1
// ---------------------------------------------------------------------------
__global__ __launch_bounds__(256) void sage_scatter1(
    const float* __restrict__ x, const int* __restrict__ ei,
    float* __restrict__ agg1, float* __restrict__ cnt, int n_edges) {
  const int lane = threadIdx.x & 31;
  const int e = (int)((blockIdx.x * (size_t)blockDim.x + threadIdx.x) >> 5);
  if (e >= n_edges) return;
  const int src = __builtin_nontemporal_load(ei + e);
  const int dst = __builtin_nontemporal_load(ei + n_edges + e);
  const float4 v = *(const float4*)(x + (size_t)src * D_FEAT + lane * 4);
  float* a = agg1 + (size_t)dst * D_FEAT + lane * 4;
  atomicAdd(a + 0, v.x);
  atomicAdd(a + 1, v.y);
  atomicAdd(a + 2, v.z);
  atomicAdd(a + 3, v.w);
  if (lane == 0) atomicAdd(cnt + dst, 1.0f);
}

// ---------------------------------------------------------------------------
// 2) layer 1 (WMMA): h = relu((agg1/cnt) @ W1l^T + x @ W1r^T + b1)
//    One wave per 16-node tile; 8 waves/block share W1l staged in 64KB LDS
//    with an XOR chunk swizzle (slot = c ^ ((2r)&63)) that makes the 32
//    lanes' ds_load_b64 bank-pair indices (c0 ^ h ^ 2q) mod 32 all distinct
//    -> conflict-free. W1r stays in WGP$ (64KB, now fits alone).
//
//    V_WMMA_F32_16X16X4_F32 fragments (ISA 7.12.2):
//    A (16x4):  lane l -> row (l&15), K = k + 2*(l>>4) + {0,1}
//    B (4x16):  lane l -> col (l&15), K = k + 2*(l>>4) + {0,1}; B[k,n]=W[n,k]
//    C/D:       lane l -> col (l&15), rows v + 8*(l>>4), v=0..7
// ---------------------------------------------------------------------------
__global__ __launch_bounds__(256) void sage_layer1_wmma(
    const float* __restrict__ x, const float* __restrict__ agg1,
    const float* __restrict__ cnt, const float* __restrict__ W1l,
    const float* __restrict__ W1r, const float* __restrict__ b1,
    float* __restrict__ h, int n_nodes) {
  __shared__ float ldsWl[D_FEAT * D_FEAT];  // 64 KB, chunk-swizzled

  // --- cooperative swizzled stage of W1l: all waves participate, then sync
  for (int idx = threadIdx.x; idx < D_FEAT * 64; idx += 256) {
    const int r = idx >> 6;   // row of W1l
    const int c = idx & 63;   // 2-dword chunk within row
    const float2 w = *(const float2*)(W1l + (size_t)r * D_FEAT + 2 * c);
    const int s = c ^ ((2 * r) & 63);
    *(float2*)(ldsWl + r * D_FEAT + 2 * s) = w;
  }
  __syncthreads();

  const int lane = threadIdx.x & 31;
  const int wave = threadIdx.x >> 5;
  const int tile = blockIdx.x * 8 + wave;
  const int n_tiles = n_nodes >> 4;
  if (tile >= n_tiles) return;  // wave-uniform: EXEC all-ones at every WMMA

  const int half = lane >> 4;
  const int q = lane & 15;
  const int rowA = tile * 16 + q;

  const float inv = 1.0f / fmaxf(cnt[rowA], 1.0f);

  v8f acc[8];
#pragma unroll
  for (int i = 0; i < 8; ++i) {
    v8f z = {};
    acc[i] = z;
  }

  const float* aggRow = agg1 + (size_t)rowA * D_FEAT + 2 * half;
  const float* xRow   = x    + (size_t)rowA * D_FEAT + 2 * half;

  // per-lane B row index and its swizzle key, per n-tile
  int nrow[8], nswz[8];
#pragma unroll
  for (int nt = 0; nt < 8; ++nt) {
    nrow[nt] = nt * 16 + q;
    nswz[nt] = (2 * nrow[nt]) & 63;
  }

  for (int k = 0; k < D_FEAT; k += 4) {
    const int c0 = (k >> 1) + half;  // 2-dword chunk index for this lane
    const float2 tm = *(const float2*)(aggRow + k);
    const float2 ts = *(const float2*)(xRow + k);

    // batch ALL fragment loads first -> scheduler can clause them and
    // count loadcnt/dscnt down across the wmma chain instead of wait-0
    float2 bl[8], br[8];
#pragma unroll
    for (int nt = 0; nt < 8; ++nt) {
      bl[nt] = *(const float2*)(ldsWl + nrow[nt] * D_FEAT + 2 * (c0 ^ nswz[nt]));
      br[nt] = *(const float2*)(W1r + (size_t)nrow[nt] * D_FEAT + k + 2 * half);
    }

    v2f a_m;  a_m.x = tm.x * inv;  a_m.y = tm.y * inv;   // mean-aggregated A
    v2f a_s;  a_s.x = ts.x;        a_s.y = ts.y;         // self-features A
#pragma unroll
    for (int nt = 0; nt < 8; ++nt) {
      v2f vbl;  vbl.x = bl[nt].x;  vbl.y = bl[nt].y;
      v2f vbr;  vbr.x = br[nt].x;  vbr.y = br[nt].y;
      acc[nt] = __builtin_amdgcn_wmma_f32_16x16x4_f32(
          false, a_m, false, vbl, (short)0, acc[nt], false, false);
      acc[nt] = __builtin_amdgcn_wmma_f32_16x16x4_f32(
          false, a_s, false, vbr, (short)0, acc[nt], false, false);
    }
  }

#pragma unroll
  for (int nt = 0; nt < 8; ++nt) {
    const int n = nrow[nt];
    const float bias = b1[n];
#pragma unroll
    for (int v = 0; v < 8; ++v) {
      const int m = v + 8 * half;
      float val = acc[nt][v] + bias;
      val = fmaxf(val, 0.0f);
      h[(size_t)(tile * 16 + m) * D_FEAT + n] = val;
    }
  }
}

// ---------------------------------------------------------------------------
// 3) per-node: p = h @ W2l^T (stored to ws), out = h @ W2r^T + b2
//    One wave per node; float4 loads; wave32 shfl_xor reduction.
// ---------------------------------------------------------------------------
__global__ __launch_bounds__(256) void sage_pq(
    const float* __restrict__ h, const float* __restrict__ W2l,
    const float* __restrict__ W2r, const float* __restrict__ b2,
    float* __restrict__ p, float* __restrict__ out, int n_nodes) {
  const int lane = threadIdx.x & 31;
  const int node = (int)((blockIdx.x * (size_t)blockDim.x + threadIdx.x) >> 5);
  if (node >= n_nodes) return;
  const float4 hv = *(const float4*)(h + (size_t)node * D_FEAT + lane * 4);
  const float4 l0 = *(const float4*)(W2l + lane * 4);
  const float4 l1 = *(const float4*)(W2l + D_FEAT + lane * 4);
  const float4 r0 = *(const float4*)(W2r + lane * 4);
  const float4 r1 = *(const float4*)(W2r + D_FEAT + lane * 4);
  float d0 = hv.x * l0.x + hv.y * l0.y + hv.z * l0.z + hv.w * l0.w;
  float d1 = hv.x * l1.x + hv.y * l1.y + hv.z * l1.z + hv.w * l1.w;
  float d2 = hv.x * r0.x + hv.y * r0.y + hv.z * r0.z + hv.w * r0.w;
  float d3 = hv.x * r1.x + hv.y * r1.y + hv.z * r1.z + hv.w * r1.w;
#pragma unroll
  for (int off = 16; off > 0; off >>= 1) {
    d0 += __shfl_xor(d0, off, 32);
    d1 += __shfl_xor(d1, off, 32);
    d2 += __shfl_xor(d2, off, 32);
    d3 += __shfl_xor(d3, off, 32);
  }
  if (lane == 0) {
    p[(size_t)node * 2 + 0] = d0;
    p[(size_t)node * 2 + 1] = d1;
    out[(size_t)node * 2 + 0] = d2 + b2[0];
    out[(size_t)node * 2 + 1] = d3 + b2[1];
  }
}

// ---------------------------------------------------------------------------
// 4) layer-2 scatter on PRE-PROJECTED features (2 floats/edge, 64x cheaper)
// ---------------------------------------------------------------------------
__global__ __launch_bounds__(256) void sage_scatter2(
    const float* __restrict__ p, const int* __restrict__ ei,
    float* __restrict__ aggp, int n_edges) {
  const int e = blockIdx.x * blockDim.x + threadIdx.x;
  if (e >= n_edges) return;
  const int src = __builtin_nontemporal_load(ei + e);
  const int dst = __builtin_nontemporal_load(ei + n_edges + e);
  const float2 pv = *(const float2*)(p + (size_t)src * 2);
  atomicAdd(aggp + (size_t)dst * 2 + 0, pv.x);
  atomicAdd(aggp + (size_t)dst * 2 + 1, pv.y);
}

// ---------------------------------------------------------------------------
// 5) finalize: out += aggp / max(cnt, 1)
// ---------------------------------------------------------------------------
__global__ void sage_finalize(const float* __restrict__ aggp,
                              const float* __restrict__ cnt,
                              float* __restrict__ out, int n_nodes) {
  const int i = blockIdx.x * blockDim.x + threadIdx.x;
  if (i >= n_nodes) return;
  const float inv = 1.0f / fmaxf(cnt[i], 1.0f);
  out[(size_t)i * 2 + 0] += aggp[(size_t)i * 2 + 0] * inv;
  out[(size_t)i * 2 + 1] += aggp[(size_t)i * 2 + 1] * inv;
}

// ---------------------------------------------------------------------------
extern "C" void kernel_launch(void* const* d_in, const int* in_sizes, int n_in,
                              void* d_out, int out_size, void* d_ws, size_t ws_size,
                              hipStream_t stream) {
  const float* x   = (const float*)d_in[0];
  const int*   ei  = (const int*)d_in[1];
  const float* W1l = (const float*)d_in[2];
  const float* W1r = (const float*)d_in[3];
  const float* b1  = (const float*)d_in[4];
  const float* W2l = (const float*)d_in[5];
  const float* W2r = (const float*)d_in[6];
  const float* b2  = (const float*)d_in[7];
  float* out = (float*)d_out;

  const int n_nodes = in_sizes[0] / D_FEAT;  // 50000
  const int n_edges = in_sizes[1] / 2;       // 800000

  // workspace layout (floats):
  //   [agg1: n*128][cnt: n][aggp: 2n]  <- zeroed each call
  //   [h: n*128][p: 2n]                <- fully overwritten
  float* ws   = (float*)d_ws;
  float* agg1 = ws;
  float* cnt  = agg1 + (size_t)n_nodes * D_FEAT;
  float* aggp = cnt + n_nodes;
  float* h    = aggp + (size_t)n_nodes * 2;
  float* p    = h + (size_t)n_nodes * D_FEAT;

  const size_t zn = (size_t)n_nodes * (D_FEAT + 3);
  sage_zero<<<(unsigned)((zn + 255) / 256), 256, 0, stream>>>(ws, zn);

  // wave per edge (8 edges per 256-thread block)
  sage_scatter1<<<(n_edges + 7) / 8, 256, 0, stream>>>(x, ei, agg1, cnt, n_edges);

  // wave per 16-node tile (8 tiles per block)
  const int n_tiles = n_nodes / 16;
  sage_layer1_wmma<<<(n_tiles + 7) / 8, 256, 0, stream>>>(x, agg1, cnt, W1l, W1r,
                                                          b1, h, n_nodes);

  // wave per node
  sage_pq<<<(n_nodes + 7) / 8, 256, 0, stream>>>(h, W2l, W2r, b2, p, out, n_nodes);

  // thread per edge (2-float payload thanks to pre-projection)
  sage_scatter2<<<(n_edges + 255) / 256, 256, 0, stream>>>(p, ei, aggp, n_edges);

  sage_finalize<<<(n_nodes + 255) / 256, 256, 0, stream>>>(aggp, cnt, out, n_nodes);
}